// MinibatchDiscrimination_81080392614176
// MI455X (gfx1250) — compile-verified
//
#include <hip/hip_runtime.h>
#include <hip/hip_bf16.h>

// Problem constants (from reference)
constexpr int IN_F  = 2048;              // IN_FEATURES (GEMM K)
constexpr int NCOL  = 2048;              // OUT_FEATURES*KERNEL_DIMS (GEMM N)
constexpr int BATCH = 256;               // GEMM M
constexpr int OUTF  = 256;               // OUT_FEATURES
constexpr int KD    = 8;                 // KERNEL_DIMS
constexpr int OUT_STRIDE = IN_F + OUTF;  // 2304

typedef __attribute__((ext_vector_type(2))) float v2f;
typedef __attribute__((ext_vector_type(8))) float v8f;

// ---------------------------------------------------------------------------
// Kernel 1: M = x @ reshape(T, [2048, 2048]) in exact f32 via
// V_WMMA_F32_16X16X4_F32.  Each wave computes a 32x32 tile (2x2 fragments),
// 4 WMMAs per K=4 step, 512 steps.
//
// A-fragment layout (ISA 7.12.2, 32-bit A 16x4): lane = r + 16*half holds
//   VGPR0 = A[row=r][k + 2*half], VGPR1 = A[row=r][k + 2*half + 1]
//   -> one contiguous b64 load per lane.
// B-fragment (4x16, rows striped across lanes, mirrored):
//   VGPR0 = B[k + 2*half][n0 + r], VGPR1 = B[k + 2*half + 1][n0 + r]
// C/D (ISA table): c[v] = C[m0 + 8*half + v][n0 + r]
// ---------------------------------------------------------------------------
__global__ __launch_bounds__(128) void gemm_wmma_f32(
    const float* __restrict__ X,   // [256, 2048]
    const float* __restrict__ T,   // [2048, 2048] row-major (= T[i][o][k])
    float* __restrict__ M)         // [256, 2048]
{
    const int lane   = threadIdx.x & 31;
    const int wave   = blockIdx.x * (blockDim.x >> 5) + (threadIdx.x >> 5);
    const int mGroup = wave & 7;     // 8 groups of 32 rows  (BATCH/32)
    const int nGroup = wave >> 3;    // 64 groups of 32 cols (NCOL/32)
    const int m0     = mGroup * 32;
    const int n0     = nGroup * 32;
    const int half   = lane >> 4;
    const int r      = lane & 15;

    const float* __restrict__ a0p = X + (m0 + r) * IN_F + 2 * half;
    const float* __restrict__ a1p = a0p + 16 * IN_F;
    const float* __restrict__ bp  = T + (2 * half) * NCOL + n0 + r;

    v8f c00 = {}, c01 = {}, c10 = {}, c11 = {};

#pragma unroll 2
    for (int k = 0; k < IN_F; k += 4) {
        v2f a0 = *(const v2f*)(a0p + k);
        v2f a1 = *(const v2f*)(a1p + k);
        const float* bk = bp + k * NCOL;
        v2f b0 = { bk[0],  bk[NCOL]      };
        v2f b1 = { bk[16], bk[NCOL + 16] };
        c00 = __builtin_amdgcn_wmma_f32_16x16x4_f32(false, a0, false, b0,
                                                    (short)0, c00, false, false);
        c01 = __builtin_amdgcn_wmma_f32_16x16x4_f32(false, a0, false, b1,
                                                    (short)0, c01, false, false);
        c10 = __builtin_amdgcn_wmma_f32_16x16x4_f32(false, a1, false, b0,
                                                    (short)0, c10, false, false);
        c11 = __builtin_amdgcn_wmma_f32_16x16x4_f32(false, a1, false, b1,
                                                    (short)0, c11, false, false);
    }

    float* o0 = M + (m0 + 8 * half) * NCOL + n0 + r;        // fragments row 0..15
    float* o1 = o0 + 16 * NCOL;                             // fragments row 16..31
#pragma unroll
    for (int v = 0; v < 8; ++v) {
        o0[v * NCOL]      = c00[v];
        o0[v * NCOL + 16] = c01[v];
        o1[v * NCOL]      = c10[v];
        o1[v * NCOL + 16] = c11[v];
    }
}

// ---------------------------------------------------------------------------
// Kernel 2: for one output feature o (blockIdx.x), stage M[:, o, 0:8] (8 KB)
// into LDS, then thread j computes  sum_i exp(-sum_k |M[i,o,k]-M[j,o,k]|) - 1.
// The i-loop reads uniform LDS addresses -> conflict-free broadcast.
// ---------------------------------------------------------------------------
__global__ __launch_bounds__(256) void pairwise_exp_l1(
    const float* __restrict__ M,   // [256, 2048]
    float* __restrict__ out)       // [256, 2304]
{
    __shared__ float4 sM[2 * BATCH];   // 256 rows x 8 floats

    const int o = blockIdx.x;          // 0..255
    const int j = threadIdx.x;         // 0..255

    const float4* mj = (const float4*)(M + j * NCOL + o * KD);
    float4 lo = mj[0];
    float4 hi = mj[1];
    sM[2 * j]     = lo;
    sM[2 * j + 1] = hi;
    __syncthreads();

    float acc = 0.0f;
#pragma unroll 4
    for (int i = 0; i < BATCH; ++i) {
        float4 a = sM[2 * i];
        float4 b = sM[2 * i + 1];
        float nrm = fabsf(a.x - lo.x) + fabsf(a.y - lo.y) +
                    fabsf(a.z - lo.z) + fabsf(a.w - lo.w) +
                    fabsf(b.x - hi.x) + fabsf(b.y - hi.y) +
                    fabsf(b.z - hi.z) + fabsf(b.w - hi.w);
        acc += __expf(-nrm);
    }
    // subtract self term exp(0) = 1
    out[j * OUT_STRIDE + IN_F + o] = acc - 1.0f;
}

// ---------------------------------------------------------------------------
// Kernel 3: copy x into the first 2048 columns of out (float4, both rows
// 16B-aligned since 2304 % 4 == 0).
// ---------------------------------------------------------------------------
__global__ __launch_bounds__(256) void copy_x(
    const float* __restrict__ x,   // [256, 2048]
    float* __restrict__ out)       // [256, 2304]
{
    int t = blockIdx.x * blockDim.x + threadIdx.x;  // 0 .. 256*512-1
    int b = t >> 9;          // row (512 float4 per row)
    int c = t & 511;         // float4 column
    ((float4*)(out + b * OUT_STRIDE))[c] = ((const float4*)(x + b * IN_F))[c];
}

extern "C" void kernel_launch(void* const* d_in, const int* in_sizes, int n_in,
                              void* d_out, int out_size, void* d_ws, size_t ws_size,
                              hipStream_t stream) {
    const float* x = (const float*)d_in[0];   // [256, 2048] f32
    const float* T = (const float*)d_in[1];   // [2048, 256, 8] f32 -> [2048,2048]
    float* out = (float*)d_out;               // [256, 2304] f32
    float* M   = (float*)d_ws;                // scratch: [256, 2048] f32 = 2 MB

    // 512 waves (8 m-groups x 64 n-groups), 4 waves per block
    gemm_wmma_f32<<<128, 128, 0, stream>>>(x, T, M);
    copy_x<<<512, 256, 0, stream>>>(x, out);
    pairwise_exp_l1<<<256, 256, 0, stream>>>(M, out);
}